// Network_10256381903586
// MI455X (gfx1250) — compile-verified
//
#include <hip/hip_runtime.h>
#include <math.h>

// ---------------------------------------------------------------------------
// Fused RHS for the population-density neuron model (N = 8,000,000).
// Memory-bound stencil: read y (64MB) once via async-to-LDS, write out (64MB).
// gfx1250: GLOBAL_LOAD_ASYNC_TO_LDS_B128/B32 + s_wait_asynccnt, wave32 shfl
// butterfly reduction, deterministic (no atomics). 4 elements/thread.
// ---------------------------------------------------------------------------

#ifndef __has_builtin
#define __has_builtin(x) 0
#endif

#if __has_builtin(__builtin_amdgcn_global_load_async_to_lds_b32)
#define HAS_ASYNC_LDS 1
#else
#define HAS_ASYNC_LDS 0
#endif

typedef int __attribute__((address_space(1))) i32_g;   // global scalar
typedef int __attribute__((address_space(3))) i32_l;   // LDS scalar
typedef __attribute__((ext_vector_type(4))) int v4i;   // int4 vector
typedef v4i __attribute__((address_space(1))) v4i_g;   // global int4
typedef v4i __attribute__((address_space(3))) v4i_l;   // LDS int4

#define TT    256                 // threads per block
#define ELEMS 4                   // elements per thread
#define TE    (TT * ELEMS)        // 1024 elements per block
#define SLOTS (TE + 8)            // LDS slots: global [base-4 .. base+1027]

// Model constants (from the reference)
#define C_GL        0.1f
#define C_EL       -5.0f
#define C_INV_CM    (1.0f/0.3f)
#define C_IEXT      0.4f
#define C_INV_DTS   2.0f          // 1/DTS, DTS = 0.5
#define C_COEF      0.4f          // 0.5*(1 - DT/DTS)
#define C_INV_SQRT3 0.57735026918962576f  // 1/(SIGMA*SQ2); SIGMA*sqrt2 = sqrt3
#define C_SQ2       1.41421356237309515f
#define C_SQ2PI     0.7978845608028654f

__device__ __forceinline__ void stage_b32(float* lds_dst, const float* g_src) {
#if HAS_ASYNC_LDS
    __builtin_amdgcn_global_load_async_to_lds_b32(
        (i32_g*)(unsigned long long)(g_src),
        (i32_l*)(unsigned int)(unsigned long long)(lds_dst), 0, 0);
#else
    *lds_dst = __ldg(g_src);
#endif
}

__device__ __forceinline__ void stage_b128(float* lds_dst, const float* g_src) {
#if HAS_ASYNC_LDS && __has_builtin(__builtin_amdgcn_global_load_async_to_lds_b128)
    __builtin_amdgcn_global_load_async_to_lds_b128(
        (v4i_g*)(unsigned long long)(g_src),
        (v4i_l*)(unsigned int)(unsigned long long)(lds_dst), 0, 0);
#elif HAS_ASYNC_LDS
    stage_b32(lds_dst + 0, g_src + 0);
    stage_b32(lds_dst + 1, g_src + 1);
    stage_b32(lds_dst + 2, g_src + 2);
    stage_b32(lds_dst + 3, g_src + 3);
#else
    *(float4*)lds_dst = *(const float4*)g_src;
#endif
}

__device__ __forceinline__ void wait_stage_then_barrier() {
#if HAS_ASYNC_LDS
#if __has_builtin(__builtin_amdgcn_s_wait_asynccnt)
    __builtin_amdgcn_s_wait_asynccnt(0);
#else
    asm volatile("s_wait_asynccnt 0" ::: "memory");
#endif
#endif
    __syncthreads();
}

// Reference _limiter simplifies: idx1&idx2 is always false, so result is m.
__device__ __forceinline__ float limiter(float a, float b) {
    return fminf(0.5f * fabsf(a + b), 2.0f * fminf(fabsf(a), fabsf(b)));
}

// H with /tau_m folded in: H = max(A*inv_tau - SQ2*dT_dt*F_T, 0)
__device__ __forceinline__ float Hfun(float V, float dvdt, float inv_tau) {
    float T  = fmaxf(-V, -1.0f) * C_INV_SQRT3;   // (VT - V), VT = 0
    float p  = 0.0061f + T*(-1.12f + T*(-0.257f + T*(-0.072f + T*(-0.0117f))));
    float A  = __expf(p);
    float dT = fminf(-dvdt * C_INV_SQRT3, 0.0f);
    float F  = C_SQ2PI * __fdividef(__expf(-T*T), 1.00000001f + erff(T));
    return fmaxf(A * inv_tau - C_SQ2 * dT * F, 0.0f);
}

__global__ void __launch_bounds__(TT)
fused_rhs(const float* __restrict__ y,     // [2N]: ro | V
          const float* __restrict__ gsyn,  // [1]
          const float* __restrict__ Isyn,  // [1]
          float* __restrict__ out,         // [2N]: dro_dt | dV_dt
          float* __restrict__ partials,    // [gridDim.x]
          int N)
{
    __shared__ __align__(16) float s_ro[SLOTS];  // s[i] <-> global base-4+i
    __shared__ __align__(16) float s_v [SLOTS];
    __shared__ float red[TT / 32];

    const int tid    = threadIdx.x;
    const int base   = blockIdx.x * TE;
    const int gstart = base - 4;
    const float* Vg  = y + N;

    // Interior blocks: every slot in range and 16B-aligned vector IO is legal.
    const bool interior = (gstart >= 0) && (base + TE + 4 <= N) && ((N & 3) == 0);

    if (interior) {
        stage_b128(&s_ro[4 * tid], &y [gstart + 4 * tid]);
        stage_b128(&s_v [4 * tid], &Vg[gstart + 4 * tid]);
        if (tid < 2) {
            const int grp = TT + tid;                     // groups 256, 257
            stage_b128(&s_ro[4 * grp], &y [gstart + 4 * grp]);
            stage_b128(&s_v [4 * grp], &Vg[gstart + 4 * grp]);
        }
    } else {
        for (int i = tid; i < SLOTS; i += TT) {
            int g = gstart + i;
            g = (g < 0) ? 0 : (g >= N ? N - 1 : g);
            stage_b32(&s_ro[i], &y [g]);
            stage_b32(&s_v [i], &Vg[g]);
        }
    }
    wait_stage_then_barrier();

    const float inv_tau = (C_GL + gsyn[0]) * C_INV_CM;   // 1/tau_m
    const float Is      = Isyn[0];
    const int   j0      = base + 4 * tid;

    // r[i] = s[4*tid + i] <-> global j0 - 4 + i ; element j0+k -> r[4+k]
    float r[9], v[9];
    {
        float4 a0 = *(const float4*)&s_ro[4 * tid];
        float4 a1 = *(const float4*)&s_ro[4 * tid + 4];
        float4 b0 = *(const float4*)&s_v [4 * tid];
        float4 b1 = *(const float4*)&s_v [4 * tid + 4];
        r[0]=a0.x; r[1]=a0.y; r[2]=a0.z; r[3]=a0.w;
        r[4]=a1.x; r[5]=a1.y; r[6]=a1.z; r[7]=a1.w; r[8]=s_ro[4 * tid + 8];
        v[0]=b0.x; v[1]=b0.y; v[2]=b0.z; v[3]=b0.w;
        v[4]=b1.x; v[5]=b1.y; v[6]=b1.z; v[7]=b1.w; v[8]=s_v [4 * tid + 8];
    }

    float sacc = 0.0f;

    if (interior) {
        float oro[ELEMS], ov[ELEMS];
        #pragma unroll
        for (int k = 0; k < ELEMS; ++k) {
            const float zm2=r[k+2], zm1=r[k+3], z0=r[k+4], zp1=r[k+5];
            const float wm2=v[k+2], wm1=v[k+3], w0=v[k+4], wp1=v[k+5];
            const float dv = (C_GL * (C_EL - w0) + C_IEXT + Is) * C_INV_CM;
            const float H  = Hfun(w0, dv, inv_tau);
            const float sr = z0 * H;
            sacc += sr;
            const float djm1=z0-zm1, djm2=zm1-zm2, dj=zp1-z0;
            oro[k] = -(djm1 + C_COEF*(limiter(dj,djm1) - limiter(djm1,djm2)))*C_INV_DTS - sr;
            const float evm1=w0-wm1, evm2=wm1-wm2, ev=wp1-w0;
            ov[k]  = -(evm1 + C_COEF*(limiter(ev,evm1) - limiter(evm1,evm2)))*C_INV_DTS + dv;
        }
        *(float4*)(out + j0)     = make_float4(oro[0], oro[1], oro[2], oro[3]);
        *(float4*)(out + N + j0) = make_float4(ov[0],  ov[1],  ov[2],  ov[3]);
    } else {
        #pragma unroll
        for (int k = 0; k < ELEMS; ++k) {
            const int j = j0 + k;
            const float zm2=r[k+2], zm1=r[k+3], z0=r[k+4], zp1=r[k+5];
            const float wm2=v[k+2], wm1=v[k+3], w0=v[k+4], wp1=v[k+5];
            const float dv = (C_GL * (C_EL - w0) + C_IEXT + Is) * C_INV_CM;
            const float H  = Hfun(w0, dv, inv_tau);
            const float sr = (j < N) ? z0 * H : 0.0f;
            sacc += sr;
            if (j >= N) continue;

            // dro_dt
            {
                const float djm1=z0-zm1, djm2=zm1-zm2, dj=zp1-z0;
                const float wj   = limiter(dj, djm1);
                const float wjm1 = (j == 1) ? 0.0f : limiter(djm1, djm2);
                if (j == 0) {
                    // out[0] = -ro[0]/DTS + firing : written by finalize
                } else if (j == N - 1) {
                    out[j] = (zm1 + C_COEF * wjm1) * C_INV_DTS - sr;
                } else {
                    out[j] = -(djm1 + C_COEF * (wj - wjm1)) * C_INV_DTS - sr;
                }
            }
            // dV_dt
            {
                if (j == 0) {
                    out[N] = 0.0f;
                } else if (j == N - 1) {
                    out[N + j] = dv;
                } else {
                    const float evm1=w0-wm1, evm2=wm1-wm2, ev=wp1-w0;
                    const float uj   = limiter(ev, evm1);
                    const float ujm1 = (j == 1) ? 0.0f : limiter(evm1, evm2);
                    out[N + j] = -(evm1 + C_COEF * (uj - ujm1)) * C_INV_DTS + dv;
                }
            }
        }
    }

    // --- deterministic firing partial: wave32 butterfly, then 8-way LDS ---
    float s = sacc;
    #pragma unroll
    for (int m = 16; m > 0; m >>= 1) s += __shfl_xor(s, m, 32);
    if ((tid & 31) == 0) red[tid >> 5] = s;
    __syncthreads();
    if (tid == 0) {
        float t = 0.0f;
        #pragma unroll
        for (int w = 0; w < TT / 32; ++w) t += red[w];   // fixed order
        partials[blockIdx.x] = t;
    }
}

// Deterministic reduction of per-block partials; writes out[0].
__global__ void __launch_bounds__(256)
finalize(const float* __restrict__ y, const float* __restrict__ partials,
         int nparts, float* __restrict__ out)
{
    __shared__ float red[256];
    const int tid = threadIdx.x;
    float s = 0.0f;
    for (int i = tid; i < nparts; i += 256) s += partials[i];  // fixed order
    red[tid] = s;
    __syncthreads();
    #pragma unroll
    for (int k = 128; k > 0; k >>= 1) {
        if (tid < k) red[tid] += red[tid + k];
        __syncthreads();
    }
    if (tid == 0) {
        // d0 = -ro[0]/DTS - src[0], src[0] = -firing
        out[0] = -y[0] * C_INV_DTS + red[0];
    }
}

extern "C" void kernel_launch(void* const* d_in, const int* in_sizes, int n_in,
                              void* d_out, int out_size, void* d_ws, size_t ws_size,
                              hipStream_t stream) {
    // inputs (setup_inputs order): t[1], y[2N], gsyn[1], Isyn[1]
    const float* y    = (const float*)d_in[1];
    const float* gsyn = (const float*)d_in[2];
    const float* Isyn = (const float*)d_in[3];
    float* out = (float*)d_out;

    const int N  = in_sizes[1] / 2;
    const int nb = (N + TE - 1) / TE;
    float* partials = (float*)d_ws;   // nb floats of scratch

    fused_rhs<<<dim3(nb), dim3(TT), 0, stream>>>(y, gsyn, Isyn, out, partials, N);
    finalize<<<dim3(1), dim3(256), 0, stream>>>(y, partials, nb, out);
}